// decoder_22565758174009
// MI455X (gfx1250) — compile-verified
//
#include <hip/hip_runtime.h>
#include <math.h>

// ---------------- problem constants ----------------
#define T_STEPS 1000
#define KDIM 32
#define D0 128
#define D1 256
#define D2 512
#define IMGH 105
#define IMGW 105
#define DT_TAU 0.01f
#define P_MIN 1e-16f

// ---------------- launch config ----------------
#define NBLK 32
#define TPB 512
#define WPB (TPB / 32)
#define NWAVE (NBLK * WPB) // 512 waves in the persistent grid

// ---------------- output layout (floats, concatenated in return order) ----------------
#define O_A0 0        // (T,32)   = z1 broadcast
#define O_A1 32000    // (T,32)   softmax(W_a0@x0+b)
#define O_A2 64000    // (T,32)   softmax(W_a1@x1+b)
#define O_U0 96000    // (T,128)  zeros
#define O_U1 224000   // (T,256)
#define O_U2 480000   // (T,512)
#define O_X0 992000   // (T,128)
#define O_X1 1120000  // (T,256)
#define O_X2 1376000  // (T,512)
#define O_PEN 1888000 // (T,2)
#define O_PXY 1890000 // (T,105,105)
#define O_PDLP 12915000 // (T,)

// ---------------- workspace layout ----------------
// bytes [0,64): grid-barrier counter (memset each launch)
// floats from ((float*)ws)+16:
#define W_XA 0          // x ping buffer (128+256+512 = 896 floats)
#define W_XB 896        // x pong buffer
#define W_L0 1792       // alpha logits layer-0 gemv (32)
#define W_L1 1824       // alpha logits layer-1 gemv (32)
#define W_IP 1856       // initial pen position (y,x)

// ---------------- wave helpers (wave32 on gfx1250) ----------------
__device__ __forceinline__ float wave_sum(float v) {
  for (int m = 16; m >= 1; m >>= 1) v += __shfl_xor(v, m, 32);
  return v;
}
__device__ __forceinline__ float wave_max(float v) {
  for (int m = 16; m >= 1; m >>= 1) v = fmaxf(v, __shfl_xor(v, m, 32));
  return v;
}

// one wave computes dot(w, x) with len a multiple of 128; result in all lanes
__device__ __forceinline__ float wave_dot(const float* __restrict__ w,
                                          const float* __restrict__ x,
                                          int len, int lane) {
  const float4* w4 = (const float4*)w;
  const float4* x4 = (const float4*)x;
  float acc = 0.f;
  int n4 = len >> 2;
  for (int s = lane; s < n4; s += 32) {
    float4 a = w4[s], b = x4[s];
    acc += a.x * b.x + a.y * b.y + a.z * b.z + a.w * b.w;
  }
  return wave_sum(acc);
}

// agent-scope grid barrier (monotonic counter, reset via hipMemsetAsync per launch)
__device__ __forceinline__ void grid_barrier(unsigned* bar, unsigned target) {
  __threadfence();      // publish this thread's global stores at agent scope
  __syncthreads();
  if (threadIdx.x == 0) {
    __hip_atomic_fetch_add(bar, 1u, __ATOMIC_ACQ_REL, __HIP_MEMORY_SCOPE_AGENT);
    while (__hip_atomic_load(bar, __ATOMIC_ACQUIRE, __HIP_MEMORY_SCOPE_AGENT) < target)
      __builtin_amdgcn_s_sleep(1);
  }
  __syncthreads();
}

// ---------------- phase 0: init ----------------
__global__ __launch_bounds__(256) void init_kernel(const unsigned char* __restrict__ data,
                                                   const float* __restrict__ z1,
                                                   const float* __restrict__ z2,
                                                   float* __restrict__ out,
                                                   float* __restrict__ wsf) {
  int i = blockIdx.x * blockDim.x + threadIdx.x;
  if (i < T_STEPS * KDIM) out[O_A0 + i] = z1[i & 31]; // alphas[0][t] == z1 for all t
  if (i < D0 + D1 + D2) wsf[W_XA + i] = (i < D0) ? z2[i] : 0.f; // x0 = [z2, 0, 0]
  if (i == 0) { // argwhere(data)[0], row-major first True (bool = 1 byte on device)
    int idx = 0;
    for (int p = 0; p < IMGH * IMGW; p++) { if (data[p] != 0) { idx = p; break; } }
    wsf[W_IP]     = (float)(idx / IMGW);
    wsf[W_IP + 1] = (float)(idx % IMGW);
  }
}

// ---------------- phase 1: sequential scan (persistent cooperative kernel) ----------------
__global__ __launch_bounds__(TPB) void scan_kernel(
    const float* __restrict__ z1, const float* __restrict__ gamma,
    const float* __restrict__ Wa0, const float* __restrict__ Wa1,
    const float* __restrict__ ba0, const float* __restrict__ ba1,
    const float* __restrict__ Wu0, const float* __restrict__ Wu1,
    const float* __restrict__ Wp, const float* __restrict__ bp,
    const float* __restrict__ A0, const float* __restrict__ A1,
    const float* __restrict__ A2,
    float* __restrict__ out, unsigned* __restrict__ bar, float* __restrict__ wsf) {
  const int lane = threadIdx.x & 31;
  const int gw = blockIdx.x * WPB + (threadIdx.x >> 5);
  unsigned epoch = 0;

  // pen recurrence lives in lane 0 of wave 0 (block-sequential, no grid sync needed)
  float pen_y = 0.f, pen_x = 0.f;
  if (gw == 0 && lane == 0) { pen_y = wsf[W_IP]; pen_x = wsf[W_IP + 1]; }

  for (int t = 0; t < T_STEPS; t++) {
    const float* xp = wsf + ((t & 1) ? W_XB : W_XA);
    float*       xn = wsf + ((t & 1) ? W_XA : W_XB);
    const float* x0 = xp;
    const float* x1 = xp + D0;
    const float* x2 = xp + D0 + D1;

    // ---- stage A: GEMVs (alpha logits + u). 832 wave-tasks ----
    for (int task = gw; task < 832; task += NWAVE) {
      if (task < 32) {
        float r = wave_dot(Wa0 + task * D0, x0, D0, lane) + ba0[task];
        if (lane == 0) wsf[W_L0 + task] = r;
      } else if (task < 64) {
        int kk = task - 32;
        float r = wave_dot(Wa1 + kk * D1, x1, D1, lane) + ba1[kk];
        if (lane == 0) wsf[W_L1 + kk] = r;
      } else if (task < 320) {
        int i = task - 64;
        float r = wave_dot(Wu0 + i * D0, x0, D0, lane);
        if (lane == 0) out[O_U1 + t * D1 + i] = r;
      } else {
        int i = task - 320;
        float r = wave_dot(Wu1 + i * D1, x1, D1, lane);
        if (lane == 0) out[O_U2 + t * D2 + i] = r;
      }
    }
    epoch++; grid_barrier(bar, epoch * NBLK); // GB1

    // ---- stage C: aAx contraction + state update. 896 rows + 2 alpha writers ----
    for (int task = gw; task < 898; task += NWAVE) {
      if (task >= 896) { // write softmax alphas to output (lane k -> alpha[k])
        int l = task - 896;
        float lg = wsf[(l == 0 ? W_L0 : W_L1) + lane];
        float m = wave_max(lg);
        float e = __expf(lg - m);
        float s = wave_sum(e);
        out[(l == 0 ? O_A1 : O_A2) + t * KDIM + lane] = e / s;
        continue;
      }
      int l, i, dim, xoff;
      const float* A; const float* xl;
      if (task < 128)      { l = 0; i = task;       dim = D0; A = A0; xl = x0; xoff = 0; }
      else if (task < 384) { l = 1; i = task - 128; dim = D1; A = A1; xl = x1; xoff = D0; }
      else                 { l = 2; i = task - 384; dim = D2; A = A2; xl = x2; xoff = D0 + D1; }

      // per-lane alpha[lane] for this layer (softmax recomputed locally: cheaper than a barrier)
      float alpha;
      if (l == 0) alpha = z1[lane];
      else {
        float lg = wsf[(l == 1 ? W_L0 : W_L1) + lane];
        float m = wave_max(lg);
        float e = __expf(lg - m);
        float s = wave_sum(e);
        alpha = e / s;
      }

      // preload x into registers (reused across all K=32 rows)
      float4 xr[4];
      const float4* x4 = (const float4*)xl;
      const int n4pl = dim >> 7; // float4's per lane: 1/2/4
      for (int s = 0; s < n4pl; s++) xr[s] = x4[lane + 32 * s];

      float acc = 0.f;
      for (int k = 0; k < KDIM; k++) {
        const float4* row4 = (const float4*)(A + (size_t)(k * dim + i) * dim);
        if (k + 1 < KDIM) // L2-resident: prefetch next k-row (global_prefetch_b8)
          __builtin_prefetch(A + (size_t)((k + 1) * dim + i) * dim, 0, 0);
        float p = 0.f;
        for (int s = 0; s < n4pl; s++) {
          float4 a = row4[lane + 32 * s];
          p += a.x * xr[s].x + a.y * xr[s].y + a.z * xr[s].z + a.w * xr[s].w;
        }
        acc += __shfl(alpha, k, 32) * p;
      }
      acc = wave_sum(acc);
      if (lane == 0) {
        float g = gamma[l];
        float xi = xl[i];
        float ui = (l == 0) ? 0.f : (l == 1 ? out[O_U1 + t * D1 + i] : out[O_U2 + t * D2 + i]);
        float xnew = xi + ((1.f - g) * acc - g * xi + ui) * DT_TAU;
        xn[xoff + i] = xnew;
        out[(l == 0 ? O_X0 : (l == 1 ? O_X1 : O_X2)) + t * dim + i] = xnew;
      }
    }
    epoch++; grid_barrier(bar, epoch * NBLK); // GB2 (x_new visible)

    // ---- stage D: pen chain on wave 0 only (overlaps next step's stage A) ----
    if (gw == 0) {
      const float* x2n = xn + D0 + D1;
      float pa0, pa1, pa2;
      {
        float p = 0.f; for (int s = lane; s < D2; s += 32) p += Wp[0 * D2 + s] * x2n[s];
        pa0 = wave_sum(p) + bp[0];
        p = 0.f;       for (int s = lane; s < D2; s += 32) p += Wp[1 * D2 + s] * x2n[s];
        pa1 = wave_sum(p) + bp[1];
        p = 0.f;       for (int s = lane; s < D2; s += 32) p += Wp[2 * D2 + s] * x2n[s];
        pa2 = wave_sum(p) + bp[2];
      }
      if (lane == 0) {
        float v = pa2; // log_sigmoid, numerically stable
        float pdlp = (v >= 0.f) ? -log1pf(__expf(-v)) : v - log1pf(__expf(v));
        out[O_PDLP + t] = pdlp;
        pen_y += pa0; pen_x += pa1;
        pen_y = 105.f / (1.f + __expf(-(pen_y - 52.5f) * (4.f / 105.f)));
        pen_x = 105.f / (1.f + __expf(-(pen_x - 52.5f) * (4.f / 105.f)));
        out[O_PEN + t * 2]     = pen_y;
        out[O_PEN + t * 2 + 1] = pen_x;
      }
    }
  }
}

// ---------------- phase 2: p_xy via WMMA rank-2 outer sum ----------------
// D[m,n] = ll_y[m]*1 + 1*ll_x[n] + pdlp  ==  [ll_y | 1] (16x4, K2..3=0) @ [[1];[ll_x];0;0] (4x16) + C
// Edge tiles are clamped inward (y0 = min(ty*16, 89)); overlapping cells recompute the same
// closed-form value, so duplicate stores are byte-identical -> no bounds checks, no exec churn.
typedef __attribute__((ext_vector_type(2))) float v2f;
typedef __attribute__((ext_vector_type(8))) float v8f;

__global__ __launch_bounds__(256) void pxy_kernel(const float* __restrict__ plv,
                                                  float* __restrict__ out,
                                                  const float* __restrict__ wsf) {
  const int wid  = (blockIdx.x * blockDim.x + threadIdx.x) >> 5; // exactly 49000 waves
  const int lane = threadIdx.x & 31;
  const int t = wid / 49;
  const int tile = wid % 49; // 7x7 tiles of 16x16 cover 105x105 (last row/col clamped)
  const int ty = tile / 7, tx = tile % 7;
  const int y0 = (ty < 6) ? ty * 16 : (IMGH - 16);
  const int x0 = (tx < 6) ? tx * 16 : (IMGW - 16);

  float py, px;
  if (t == 0) { py = wsf[W_IP]; px = wsf[W_IP + 1]; }
  else        { py = out[O_PEN + (t - 1) * 2]; px = out[O_PEN + (t - 1) * 2 + 1]; }
  const float pdlp = out[O_PDLP + t];
  const float inv_var = 1.0f / (__expf(plv[0]) + P_MIN); // exp(-log(exp(plv)+P_MIN))

  // A 16x4 f32: lanes 0-15 {v0=K0=ll_y[m], v1=K1=1}; lanes 16-31 {K2,K3}=0
  // B 4x16 f32: lanes 0-15 {v0=K0 row=1, v1=K1 row=ll_x[n]}; lanes 16-31 {K2,K3 rows}=0
  v2f a, b;
  if (lane < 16) {
    float gy = (float)(y0 + lane) + 0.5f;
    float gx = (float)(x0 + lane) + 0.5f;
    float dy = py - gy, dx = px - gx;
    a.x = -0.5f * dy * dy * inv_var; a.y = 1.0f;
    b.x = 1.0f;                      b.y = -0.5f * dx * dx * inv_var;
  } else { a.x = 0.f; a.y = 0.f; b.x = 0.f; b.y = 0.f; }

  v8f c;
  for (int r = 0; r < 8; r++) c[r] = pdlp;
  v8f d;
#if __has_builtin(__builtin_amdgcn_wmma_f32_16x16x4_f32)
  d = __builtin_amdgcn_wmma_f32_16x16x4_f32(false, a, false, b, (short)0, c,
                                            false, false);
#else
  for (int r = 0; r < 8; r++) { // VALU fallback
    int m = r + ((lane >> 4) << 3), n = lane & 15;
    float gy = (float)(y0 + m) + 0.5f, gx = (float)(x0 + n) + 0.5f;
    float dy = py - gy, dx = px - gx;
    d[r] = -0.5f * dy * dy * inv_var - 0.5f * dx * dx * inv_var + pdlp;
  }
#endif

  // C/D layout: lane<16 -> (M=r, N=lane); lane>=16 -> (M=r+8, N=lane-16).
  // Per-lane base pointer; row r advances by IMGW floats (constant immediate offsets).
  const int hi = lane >> 4;        // 0: rows 0-7, 1: rows 8-15
  const int n  = lane & 15;
  float* p = out + O_PXY + (size_t)t * (IMGH * IMGW) + (size_t)(y0 + hi * 8) * IMGW + (x0 + n);
#pragma unroll
  for (int r = 0; r < 8; r++)
    p[r * IMGW] = __expf(d[r]) * (1.0f - 2e-16f) + P_MIN;
}

// ---------------- host entry ----------------
extern "C" void kernel_launch(void* const* d_in, const int* in_sizes, int n_in,
                              void* d_out, int out_size, void* d_ws, size_t ws_size,
                              hipStream_t stream) {
  (void)in_sizes; (void)n_in; (void)out_size; (void)ws_size;
  const unsigned char* data = (const unsigned char*)d_in[0];
  const float* z1  = (const float*)d_in[1];
  const float* z2  = (const float*)d_in[2];
  const float* gm  = (const float*)d_in[3];
  const float* plv = (const float*)d_in[4];
  const float* Wa0 = (const float*)d_in[5];
  const float* Wa1 = (const float*)d_in[6];
  const float* ba0 = (const float*)d_in[7];
  const float* ba1 = (const float*)d_in[8];
  const float* Wu0 = (const float*)d_in[9];
  const float* Wu1 = (const float*)d_in[10];
  const float* Wp  = (const float*)d_in[11];
  const float* bp  = (const float*)d_in[12];
  const float* A0  = (const float*)d_in[13];
  const float* A1  = (const float*)d_in[14];
  const float* A2  = (const float*)d_in[15];
  float* out = (float*)d_out;
  unsigned* bar = (unsigned*)d_ws;
  float* wsf = (float*)d_ws + 16;

  // capture-safe per-launch resets (harness does not re-initialize between replays)
  hipMemsetAsync(d_ws, 0, 64, stream);                                  // barrier counter
  hipMemsetAsync(out + O_U0, 0, (size_t)T_STEPS * D0 * sizeof(float), stream); // u[0] == 0

  init_kernel<<<(T_STEPS * KDIM + 255) / 256, 256, 0, stream>>>(data, z1, z2, out, wsf);
  scan_kernel<<<NBLK, TPB, 0, stream>>>(z1, gm, Wa0, Wa1, ba0, ba1, Wu0, Wu1,
                                        Wp, bp, A0, A1, A2, out, bar, wsf);
  pxy_kernel<<<6125, 256, 0, stream>>>(plv, out, wsf); // exactly 49000 full waves
}